// BertEncoder_403726926494
// MI455X (gfx1250) — compile-verified
//
#include <hip/hip_runtime.h>

typedef __attribute__((ext_vector_type(2))) float v2f;
typedef __attribute__((ext_vector_type(8))) float v8f;

#define B_ 64
#define L_ 512
#define H_ 768
#define D_ 16
#define S_ 64
#define Q_ 32
#define NEGV (-900000.0f)

__device__ __forceinline__ int clipL(int v) {
    v = v < 0 ? 0 : v;
    return v > (L_ - 1) ? (L_ - 1) : v;
}

// -------- doc pooling: one workgroup per (b,d), 256 threads = 8 wave32 -----
__global__ __launch_bounds__(256) void doc_pool_kernel(
    const float* __restrict__ hidden, const float* __restrict__ Wd,
    const float* __restrict__ bd, const int* __restrict__ qlen,
    const int* __restrict__ slens, float* __restrict__ out)
{
    __shared__ __align__(16) float lds_w[H_];
    __shared__ __align__(16) float lds_alpha[S_];
    __shared__ int lds_row[S_];

    const int bd_id = blockIdx.x;
    const int b = bd_id / D_;
    const int d = bd_id % D_;
    const int tid = threadIdx.x;
    const int wave = tid >> 5;
    const int lane = tid & 31;

    // offset = qlen[b] + 2 + cumsum(seq_lens)[d] - seq_lens[d]
    int offset = qlen[b] + 2;
    for (int j = 0; j < d; ++j) offset += slens[b * D_ + j];
    const int seq_len = slens[b * D_ + d];

    for (int i = tid; i < H_; i += 256) lds_w[i] = Wd[i];
    if (tid < S_) lds_row[tid] = clipL(offset + tid);
    __syncthreads();

    // ---- phase A: alpha[s] = sent[s,:] . W_doc via v_wmma_f32_16x16x4_f32
    // A: 16 sentence rows x 4 h; B: W_doc replicated over all 16 N columns.
    if (wave < 4) {
        const int m  = lane & 15;          // M = row within 16-block
        const int s  = wave * 16 + m;
        const int kk = (lane >> 4) * 2;    // K-pair owned by this half-wave
        const float* row = hidden + ((size_t)b * L_ + lds_row[s]) * (size_t)H_;
        v8f c = {};
        for (int h0 = 0; h0 < H_; h0 += 4) {
            v2f a = *(const v2f*)(row + h0 + kk);
            v2f w = *(const v2f*)(lds_w + h0 + kk);
            c = __builtin_amdgcn_wmma_f32_16x16x4_f32(false, a, false, w,
                                                      (short)0, c, false, false);
        }
        // D layout: VGPR r = M=r (lanes 0-15) / M=8+r (lanes 16-31); all N equal
        if (lane == 0 || lane == 16) {
            const int base = wave * 16 + (lane == 16 ? 8 : 0);
            const float bias = bd[0];
#pragma unroll
            for (int r = 0; r < 8; ++r) {
                const int sidx = base + r;
                const float v = c[r] + bias;
                lds_alpha[sidx] = (sidx < seq_len) ? v : NEGV;
            }
        }
    }
    __syncthreads();

    // ---- softmax over S (redundant scan per thread; 64 LDS reads, cheap)
    float amax = -3.4e38f;
    for (int i = 0; i < S_; ++i) amax = fmaxf(amax, lds_alpha[i]);
    float sum = 0.f;
    for (int i = 0; i < S_; ++i) sum += __expf(lds_alpha[i] - amax);
    const float myv = (tid < S_) ? lds_alpha[tid] : 0.f;
    __syncthreads();
    if (tid < S_) {
        const float a = __expf(myv - amax) / sum;
        lds_alpha[tid] = (tid < seq_len) ? a : 0.f;   // mimics zeroed sent
    }
    __syncthreads();

    // ---- phase B: out[h] = sum_s alpha[s]*sent[s,h] via WMMA (N = 16 h cols)
    const int n  = lane & 15;
    const int kk = (lane >> 4) * 2;
    float* outrow = out + (size_t)bd_id * H_;
    for (int t = wave; t < H_ / 16; t += 8) {
        const int hcol = t * 16 + n;
        v8f c = {};
        for (int k0 = 0; k0 < S_; k0 += 4) {
            v2f a = *(const v2f*)(lds_alpha + k0 + kk);   // same alpha for all M
            v2f bv;
            bv.x = hidden[((size_t)b * L_ + lds_row[k0 + kk])     * (size_t)H_ + hcol];
            bv.y = hidden[((size_t)b * L_ + lds_row[k0 + kk + 1]) * (size_t)H_ + hcol];
            if (k0 + 4 < S_)  // gfx1250 global_prefetch_b8 path (L2-hot anyway)
                __builtin_prefetch(&hidden[((size_t)b * L_ + lds_row[k0 + 4 + kk]) * (size_t)H_ + hcol], 0, 0);
            c = __builtin_amdgcn_wmma_f32_16x16x4_f32(false, a, false, bv,
                                                      (short)0, c, false, false);
        }
        if (lane < 16) outrow[t * 16 + lane] = c[0];   // all D rows equal; row M=0
    }
}

// -------- query pooling: one workgroup per b ------------------------------
__global__ __launch_bounds__(256) void query_pool_kernel(
    const float* __restrict__ hidden, const float* __restrict__ Wq,
    const float* __restrict__ bq, const int* __restrict__ qlen,
    float* __restrict__ outq)
{
    __shared__ __align__(16) float lds_w[H_];
    __shared__ __align__(16) float lds_alpha[Q_];

    const int b = blockIdx.x;
    const int tid = threadIdx.x;
    const int wave = tid >> 5;
    const int lane = tid & 31;
    const int ql = qlen[b];

    for (int i = tid; i < H_; i += 256) lds_w[i] = Wq[i];
    __syncthreads();

    if (wave < 2) {
        const int m  = lane & 15;
        const int q  = wave * 16 + m;
        const int kk = (lane >> 4) * 2;
        const float* row = hidden + ((size_t)b * L_ + 1 + q) * (size_t)H_;
        v8f c = {};
        for (int h0 = 0; h0 < H_; h0 += 4) {
            v2f a = *(const v2f*)(row + h0 + kk);
            v2f w = *(const v2f*)(lds_w + h0 + kk);
            c = __builtin_amdgcn_wmma_f32_16x16x4_f32(false, a, false, w,
                                                      (short)0, c, false, false);
        }
        if (lane == 0 || lane == 16) {
            const int base = wave * 16 + (lane == 16 ? 8 : 0);
            const float bias = bq[0];
#pragma unroll
            for (int r = 0; r < 8; ++r) {
                const int qi = base + r;
                const float v = c[r] + bias;
                lds_alpha[qi] = (qi < ql) ? v : NEGV;
            }
        }
    }
    __syncthreads();

    float amax = -3.4e38f;
    for (int i = 0; i < Q_; ++i) amax = fmaxf(amax, lds_alpha[i]);
    float sum = 0.f;
    for (int i = 0; i < Q_; ++i) sum += __expf(lds_alpha[i] - amax);
    const float myv = (tid < Q_) ? lds_alpha[tid] : 0.f;
    __syncthreads();
    if (tid < Q_) {
        const float a = __expf(myv - amax) / sum;
        lds_alpha[tid] = (tid < ql) ? a : 0.f;
    }
    __syncthreads();

    const int n  = lane & 15;
    const int kk = (lane >> 4) * 2;
    for (int t = wave; t < H_ / 16; t += 8) {
        const int hcol = t * 16 + n;
        v8f c = {};
        for (int k0 = 0; k0 < Q_; k0 += 4) {
            v2f a = *(const v2f*)(lds_alpha + k0 + kk);
            v2f bv;
            bv.x = hidden[((size_t)b * L_ + 1 + k0 + kk) * (size_t)H_ + hcol];
            bv.y = hidden[((size_t)b * L_ + 2 + k0 + kk) * (size_t)H_ + hcol];
            c = __builtin_amdgcn_wmma_f32_16x16x4_f32(false, a, false, bv,
                                                      (short)0, c, false, false);
        }
        if (lane < 16) {
            const float v = c[0];
#pragma unroll
            for (int dd = 0; dd < D_; ++dd)
                outq[((size_t)(b * D_ + dd)) * (size_t)H_ + t * 16 + lane] = v;
        }
    }
}

extern "C" void kernel_launch(void* const* d_in, const int* in_sizes, int n_in,
                              void* d_out, int out_size, void* d_ws, size_t ws_size,
                              hipStream_t stream) {
    (void)in_sizes; (void)n_in; (void)out_size; (void)d_ws; (void)ws_size;
    const float* hidden = (const float*)d_in[0];
    const float* Wd     = (const float*)d_in[1];
    const float* bd     = (const float*)d_in[2];
    const float* Wq     = (const float*)d_in[3];
    const float* bq     = (const float*)d_in[4];
    const int*   ql     = (const int*)d_in[5];
    const int*   sl     = (const int*)d_in[6];
    float* out = (float*)d_out;

    doc_pool_kernel<<<B_ * D_, 256, 0, stream>>>(hidden, Wd, bd, ql, sl, out);
    query_pool_kernel<<<B_, 256, 0, stream>>>(hidden, Wq, bq, ql,
                                              out + (size_t)B_ * D_ * H_);
}